// Point_Conv_59296318489053
// MI455X (gfx1250) — compile-verified
//
#include <hip/hip_runtime.h>
#include <hip/hip_bf16.h>

typedef __attribute__((ext_vector_type(2))) float v2f;
typedef __attribute__((ext_vector_type(8))) float v8f;

#define NPTS   8192
#define NCTR   2048   // per batch
#define NBATCH 4
#define NSAMP  32
#define CIN    67
#define COUT   128
#define KTOT   (NSAMP * CIN)   // 2144
#define KCH    536             // K chunk (multiple of 4), 4 chunks
#define ROWP   540             // padded LDS row stride (conflict-free for 16 rows)

// ---------------- FPS: one workgroup per batch, sequential argmax chain ------
__global__ __launch_bounds__(1024) void fps_kernel(const float* __restrict__ points,
                                                   float* __restrict__ new_xyz) {
    const int b = blockIdx.x;
    const float* pts = points + (size_t)b * NPTS * 3;
    float* nxz = new_xyz + (size_t)b * NCTR * 3;
    const int t = threadIdx.x;
    const int lane = t & 31, wid = t >> 5;

    float px[8], py[8], pz[8], dist[8];
#pragma unroll
    for (int j = 0; j < 8; ++j) {
        int p = t + j * 1024;
        px[j] = pts[p * 3 + 0];
        py[j] = pts[p * 3 + 1];
        pz[j] = pts[p * 3 + 2];
        dist[j] = 1.0e10f;
    }

    __shared__ float s_last[3];
    __shared__ float s_val[32];
    __shared__ int   s_idx[32];

    if (t == 0) {
        s_last[0] = pts[0]; s_last[1] = pts[1]; s_last[2] = pts[2];
        nxz[0] = pts[0]; nxz[1] = pts[1]; nxz[2] = pts[2];   // inds[0] == 0
    }
    __syncthreads();

    for (int k = 1; k < NCTR; ++k) {
        const float lx = s_last[0], ly = s_last[1], lz = s_last[2];
        float bv = -1.0f; int bi = 0;
#pragma unroll
        for (int j = 0; j < 8; ++j) {
            float dx = px[j] - lx, dy = py[j] - ly, dz = pz[j] - lz;
            float d = dx * dx + dy * dy + dz * dz;
            d = fminf(dist[j], d);
            dist[j] = d;
            int p = t + j * 1024;
            if (d > bv) { bv = d; bi = p; }   // j ascending => first-max kept
        }
        // wave32 argmax reduce, tie -> smaller index (jnp.argmax semantics)
#pragma unroll
        for (int off = 16; off > 0; off >>= 1) {
            float ov = __shfl_down(bv, off);
            int   oi = __shfl_down(bi, off);
            if (ov > bv || (ov == bv && oi < bi)) { bv = ov; bi = oi; }
        }
        if (lane == 0) { s_val[wid] = bv; s_idx[wid] = bi; }
        __syncthreads();
        if (wid == 0) {
            bv = s_val[lane]; bi = s_idx[lane];
#pragma unroll
            for (int off = 16; off > 0; off >>= 1) {
                float ov = __shfl_down(bv, off);
                int   oi = __shfl_down(bi, off);
                if (ov > bv || (ov == bv && oi < bi)) { bv = ov; bi = oi; }
            }
            if (lane == 0) {
                float x = pts[bi * 3], y = pts[bi * 3 + 1], z = pts[bi * 3 + 2];
                s_last[0] = x; s_last[1] = y; s_last[2] = z;
                nxz[k * 3 + 0] = x; nxz[k * 3 + 1] = y; nxz[k * 3 + 2] = z;
            }
        }
        __syncthreads();
    }
}

// ---------------- Ball query: one thread per center, ascending-index scan ----
__global__ void ballq_kernel(const float* __restrict__ points,
                             const float* __restrict__ new_xyz,
                             int* __restrict__ idxbuf) {
    int g = blockIdx.x * blockDim.x + threadIdx.x;   // 0..8191
    if (g >= NBATCH * NCTR) return;
    const int b = g >> 11;
    const float* pts = points + (size_t)b * NPTS * 3;
    const float cx = new_xyz[g * 3 + 0];
    const float cy = new_xyz[g * 3 + 1];
    const float cz = new_xyz[g * 3 + 2];
    const float r  = 0.2f;
    const float r2 = r * r;
    int* out = idxbuf + (size_t)g * NSAMP;
    int cnt = 0, first = 0;
    for (int i = 0; i < NPTS; ++i) {
        float dx = pts[i * 3 + 0] - cx;
        float dy = pts[i * 3 + 1] - cy;
        float dz = pts[i * 3 + 2] - cz;
        float d2 = dx * dx + dy * dy + dz * dz;
        if (d2 < r2) {
            if (cnt == 0) first = i;
            out[cnt] = i;
            if (++cnt >= NSAMP) break;
        }
    }
    for (int j = cnt; j < NSAMP; ++j) out[j] = first;   // pad with first hit
}

// ---------------- Weight transpose: W[o][c][s] -> Wt[k=s*67+c][o] ------------
__global__ void wtr_kernel(const float* __restrict__ w, float* __restrict__ wt) {
    int e = blockIdx.x * blockDim.x + threadIdx.x;
    if (e >= KTOT * COUT) return;
    int k = e >> 7, o = e & (COUT - 1);
    int s = k / CIN, c = k % CIN;
    wt[k * COUT + o] = w[(o * CIN + c) * NSAMP + s];
}

// ---------------- Gather + GEMM via V_WMMA_F32_16X16X4_F32 -------------------
// Block = 128 threads (4 waves). Block owns 16 centers; wave w owns outputs
// [w*32, w*32+32) as two 16x16 WMMA tiles. A (g-matrix) staged in LDS chunks.
__global__ __launch_bounds__(128) void conv_kernel(const float* __restrict__ points,
                                                   const float* __restrict__ features,
                                                   const float* __restrict__ wt,
                                                   const float* __restrict__ new_xyz,
                                                   const int* __restrict__ idxbuf,
                                                   float* __restrict__ out) {
    __shared__ float gA[16 * ROWP];

    const int cb = blockIdx.x * 16;          // center base (all share batch b)
    const int b  = cb >> 11;
    const float* pts  = points   + (size_t)b * NPTS * 3;
    const float* feat = features + (size_t)b * NPTS * 64;

    const int t = threadIdx.x;
    const int lane = t & 31, wave = t >> 5;
    const int mrow  = lane & 15;
    const int khalf = lane >> 4;
    const int o0 = wave * 32 + (lane & 15);
    const int o1 = o0 + 16;

    v8f acc0 = {}; v8f acc1 = {};

    for (int ch = 0; ch < 4; ++ch) {
        __syncthreads();
        // Gather this K-chunk of the 16x2144 g-matrix into LDS.
        for (int e = t; e < 16 * KCH; e += 128) {
            int m  = e / KCH;
            int kc = e % KCH;
            int kg = ch * KCH + kc;
            int s  = kg / CIN, c = kg % CIN;
            int g  = cb + m;
            int pid = idxbuf[(size_t)g * NSAMP + s];
            float val;
            if (c < 3) val = pts[pid * 3 + c] - new_xyz[g * 3 + c];
            else       val = feat[pid * 64 + (c - 3)];
            gA[m * ROWP + kc] = val;
        }
        __syncthreads();

        for (int kc = 0; kc < KCH; kc += 4) {
            const int kk = kc + 2 * khalf;          // A: V0={K0,K2}, V1={K1,K3}
            v2f a;
            a.x = gA[mrow * ROWP + kk];
            a.y = gA[mrow * ROWP + kk + 1];
            const int kgk = ch * KCH + kk;
            const float* wrow = wt + (size_t)kgk * COUT;
            v2f b0; b0.x = wrow[o0]; b0.y = wrow[COUT + o0];
            v2f b1; b1.x = wrow[o1]; b1.y = wrow[COUT + o1];
            acc0 = __builtin_amdgcn_wmma_f32_16x16x4_f32(false, a, false, b0,
                                                         (short)0, acc0, false, false);
            acc1 = __builtin_amdgcn_wmma_f32_16x16x4_f32(false, a, false, b1,
                                                         (short)0, acc1, false, false);
        }
    }

    // D layout: VGPR r -> M = r + 8*khalf, N = lane&15
    const int m0 = 8 * khalf;
#pragma unroll
    for (int r = 0; r < 8; ++r) {
        int g = cb + m0 + r;
        out[(size_t)g * COUT + o0] = acc0[r];
        out[(size_t)g * COUT + o1] = acc1[r];
    }
}

extern "C" void kernel_launch(void* const* d_in, const int* in_sizes, int n_in,
                              void* d_out, int out_size, void* d_ws, size_t ws_size,
                              hipStream_t stream) {
    const float* points   = (const float*)d_in[0];   // (4,8192,3)
    const float* features = (const float*)d_in[1];   // (4,8192,64)
    const float* weight   = (const float*)d_in[2];   // (128,67,1,32)

    float* out      = (float*)d_out;
    float* new_xyz  = out;                            // (4,2048,3)
    float* conv     = out + (size_t)NBATCH * NCTR * 3;  // (4,2048,128)

    int*   idxbuf = (int*)d_ws;                                   // 8192*32 ints
    float* wtbuf  = (float*)((char*)d_ws +
                             (size_t)NBATCH * NCTR * NSAMP * sizeof(int)); // 2144*128 f32

    fps_kernel<<<NBATCH, 1024, 0, stream>>>(points, new_xyz);
    wtr_kernel<<<(KTOT * COUT + 255) / 256, 256, 0, stream>>>(weight, wtbuf);
    ballq_kernel<<<(NBATCH * NCTR) / 256, 256, 0, stream>>>(points, new_xyz, idxbuf);
    conv_kernel<<<(NBATCH * NCTR) / 16, 128, 0, stream>>>(points, features, wtbuf,
                                                          new_xyz, idxbuf, conv);
}